// GIN_73710228734579
// MI455X (gfx1250) — compile-verified
//
#include <hip/hip_runtime.h>
#include <hip/hip_bf16.h>

// ---------------------------------------------------------------------------
// GIN forward on MI455X (gfx1250, wave32).
//   h1 = relu((x  + scatter_add(x[src]->dst))  @ W1 + b1)
//   h2 = relu((h1 + scatter_add(h1[src]->dst)) @ W2 + b2)
//   out = (segment_mean(h2, batch)) @ W3 + b3
// GEMMs use V_WMMA_F32_16X16X4_F32 (exact fp32) — 64 WMMAs per 16x64 strip.
// Weights staged in LDS with K-pair interleaving so each B fragment is one
// aligned ds_load_b64 directly into the WMMA source register pair.
// ---------------------------------------------------------------------------

#define FEAT 64

typedef float v2f __attribute__((ext_vector_type(2)));
typedef float v8f __attribute__((ext_vector_type(8)));

// ---------------------------------------------------------------------------
// Edge scatter-add: agg[dst[e]][f] += h[src[e]][f]
// One thread per (edge, feature); 64 consecutive lanes read one source row
// coalesced, atomics land in L2 (node state << 192MB L2).
// ---------------------------------------------------------------------------
__global__ __launch_bounds__(256) void gin_scatter(
    const float* __restrict__ h, const int* __restrict__ src,
    const int* __restrict__ dst, float* __restrict__ agg, long long total)
{
    long long idx = (long long)blockIdx.x * blockDim.x + threadIdx.x;
    if (idx >= total) return;
    int e = (int)(idx >> 6);
    int f = (int)(idx & 63);
    int s = src[e];
    int d = dst[e];
    atomicAdd(&agg[(size_t)d * FEAT + f], h[(size_t)s * FEAT + f]);
}

// ---------------------------------------------------------------------------
// Fused (xin + agg) @ W + b, optional ReLU, via V_WMMA_F32_16X16X4_F32.
// Block = 256 threads = 8 waves. Each wave owns one 16-row strip, producing
// a 16x64 output tile as four 16x16 f32 accumulators.
//
// fp32 WMMA fragment layouts (ISA 7.12.2):
//   A 16x4:  lanes 0-15 -> M=lane, v2f = {K0,K1};  lanes 16-31 -> {K2,K3}
//   B 4x16:  lanes 0-15 -> N=lane, v2f = {K0,K1};  lanes 16-31 -> {K2,K3}
//   C 16x16: vgpr r -> M=r (lanes 0-15) / M=8+r (lanes 16-31), N = lane&15
//
// LDS weight layout: W[k][n] at sW[(k>>1)*128 + n*2 + (k&1)], so the pair
// {W[kk][n], W[kk+1][n]} (kk even) is contiguous & 8B-aligned -> ds_load_b64.
// ---------------------------------------------------------------------------
__global__ __launch_bounds__(256) void gin_wmma_gemm(
    const float* __restrict__ xin, const float* __restrict__ agg,
    const float* __restrict__ W, const float* __restrict__ bias,
    float* __restrict__ out, int n_rows, int n_tiles, int do_relu)
{
    __shared__ float sW[FEAT * FEAT];   // K-pair interleaved
    __shared__ float sB[FEAT];

    for (int i = threadIdx.x; i < FEAT * FEAT; i += 256) {
        int k = i >> 6;        // 0..63
        int n = i & 63;
        sW[(k >> 1) * 128 + n * 2 + (k & 1)] = W[i];
    }
    if (threadIdx.x < FEAT) sB[threadIdx.x] = bias[threadIdx.x];
    __syncthreads();

    const int wave = threadIdx.x >> 5;   // 0..7
    const int lane = threadIdx.x & 31;
    const int tile = blockIdx.x * 8 + wave;
    if (tile >= n_tiles) return;         // wave-uniform: EXEC stays all-ones

    const int row0  = tile * 16;
    const int nlo   = lane & 15;         // column within 16-wide tile / A row idx
    const int khalf = (lane >> 4) << 1;  // 0 for lanes 0-15, 2 for lanes 16-31

    int arow = row0 + nlo;
    if (arow >= n_rows) arow = n_rows - 1;          // clamp (keeps EXEC full)
    const float* __restrict__ xr = xin + (size_t)arow * FEAT;
    const float* __restrict__ gr = agg + (size_t)arow * FEAT;

    v8f acc0 = {}, acc1 = {}, acc2 = {}, acc3 = {};

#pragma unroll
    for (int k0 = 0; k0 < FEAT; k0 += 4) {
        const int kk = k0 + khalf;                    // even
        v2f a;
        a.x = xr[kk]     + gr[kk];
        a.y = xr[kk + 1] + gr[kk + 1];

        const float* wrow = &sW[(kk >> 1) * 128];     // 64 n-pairs for this K-pair
        v2f b0 = *(const v2f*)&wrow[(nlo)      * 2];
        v2f b1 = *(const v2f*)&wrow[(16 + nlo) * 2];
        v2f b2 = *(const v2f*)&wrow[(32 + nlo) * 2];
        v2f b3 = *(const v2f*)&wrow[(48 + nlo) * 2];

        acc0 = __builtin_amdgcn_wmma_f32_16x16x4_f32(false, a, false, b0, (short)0, acc0, false, false);
        acc1 = __builtin_amdgcn_wmma_f32_16x16x4_f32(false, a, false, b1, (short)0, acc1, false, false);
        acc2 = __builtin_amdgcn_wmma_f32_16x16x4_f32(false, a, false, b2, (short)0, acc2, false, false);
        acc3 = __builtin_amdgcn_wmma_f32_16x16x4_f32(false, a, false, b3, (short)0, acc3, false, false);
    }

    // Epilogue: bias + (optional) ReLU, guarded store (WMMAs are done; EXEC
    // divergence is fine here).
    const int mbase = row0 + ((lane >> 4) << 3);
    const float bb0 = sB[nlo], bb1 = sB[16 + nlo], bb2 = sB[32 + nlo], bb3 = sB[48 + nlo];
#pragma unroll
    for (int r = 0; r < 8; ++r) {
        const int m = mbase + r;
        if (m < n_rows) {
            float v0 = acc0[r] + bb0;
            float v1 = acc1[r] + bb1;
            float v2 = acc2[r] + bb2;
            float v3 = acc3[r] + bb3;
            if (do_relu) {
                v0 = v0 > 0.f ? v0 : 0.f;
                v1 = v1 > 0.f ? v1 : 0.f;
                v2 = v2 > 0.f ? v2 : 0.f;
                v3 = v3 > 0.f ? v3 : 0.f;
            }
            float* o = out + (size_t)m * FEAT + nlo;
            o[0]  = v0;
            o[16] = v1;
            o[32] = v2;
            o[48] = v3;
        }
    }
}

// ---------------------------------------------------------------------------
// Segment-sum pooling: sums[batch[n]][f] += h2[n][f]; cnt[batch[n]] += 1
// ---------------------------------------------------------------------------
__global__ __launch_bounds__(256) void gin_pool(
    const float* __restrict__ h2, const int* __restrict__ batch,
    float* __restrict__ sums, float* __restrict__ cnt, long long total)
{
    long long idx = (long long)blockIdx.x * blockDim.x + threadIdx.x;
    if (idx >= total) return;
    int n = (int)(idx >> 6);
    int f = (int)(idx & 63);
    int g = batch[n];
    atomicAdd(&sums[(size_t)g * FEAT + f], h2[(size_t)n * FEAT + f]);
    if (f == 0) atomicAdd(&cnt[g], 1.0f);
}

// ---------------------------------------------------------------------------
// Final classifier: out[g][c] = (sums[g]/max(cnt[g],1)) . W3[:,c] + b3[c]
// 500x10 outputs — scalar is the right tool (matrix pipes add nothing here).
// ---------------------------------------------------------------------------
__global__ __launch_bounds__(256) void gin_head(
    const float* __restrict__ sums, const float* __restrict__ cnt,
    const float* __restrict__ W3, const float* __restrict__ b3,
    float* __restrict__ out, int n_graphs, int n_classes)
{
    int idx = blockIdx.x * blockDim.x + threadIdx.x;
    if (idx >= n_graphs * n_classes) return;
    int g = idx / n_classes;
    int c = idx - g * n_classes;
    float d = cnt[g];
    d = d < 1.0f ? 1.0f : d;
    float inv = 1.0f / d;
    float acc = b3[c];
    const float* sg = sums + (size_t)g * FEAT;
#pragma unroll 8
    for (int k = 0; k < FEAT; ++k) acc = fmaf(sg[k] * inv, W3[k * n_classes + c], acc);
    out[idx] = acc;
}

// ---------------------------------------------------------------------------
// Launcher
// ---------------------------------------------------------------------------
extern "C" void kernel_launch(void* const* d_in, const int* in_sizes, int n_in,
                              void* d_out, int out_size, void* d_ws, size_t ws_size,
                              hipStream_t stream) {
    const float* x     = (const float*)d_in[0];
    const int*   eidx  = (const int*)d_in[1];
    const int*   batch = (const int*)d_in[2];
    const float* W1    = (const float*)d_in[3];
    const float* b1    = (const float*)d_in[4];
    const float* W2    = (const float*)d_in[5];
    const float* b2    = (const float*)d_in[6];
    const float* W3    = (const float*)d_in[7];
    const float* b3    = (const float*)d_in[8];
    float* out = (float*)d_out;

    const int n_nodes = in_sizes[0] / FEAT;
    const int n_edges = in_sizes[1] / 2;
    const int n_cls   = in_sizes[8];              // b3 length = N_CLASSES
    const int n_grp   = out_size / n_cls;         // number of graphs
    (void)n_in; (void)ws_size;

    const int* src = eidx;
    const int* dst = eidx + n_edges;

    // Workspace layout
    float* agg  = (float*)d_ws;                          // n_nodes*64
    float* h1   = agg  + (size_t)n_nodes * FEAT;         // n_nodes*64
    float* h2   = h1   + (size_t)n_nodes * FEAT;         // n_nodes*64
    float* sums = h2   + (size_t)n_nodes * FEAT;         // n_grp*64
    float* cnt  = sums + (size_t)n_grp * FEAT;           // n_grp

    const size_t node_bytes = (size_t)n_nodes * FEAT * sizeof(float);
    const size_t pool_bytes = ((size_t)n_grp * FEAT + n_grp) * sizeof(float);

    const long long edge_work = (long long)n_edges * FEAT;
    const int scat_blocks  = (int)((edge_work + 255) / 256);
    const int n_tiles      = (n_nodes + 15) / 16;
    const int gemm_blocks  = (n_tiles + 7) / 8;
    const long long node_work = (long long)n_nodes * FEAT;
    const int pool_blocks  = (int)((node_work + 255) / 256);
    const int head_blocks  = (n_grp * n_cls + 255) / 256;

    // Layer 1
    hipMemsetAsync(agg, 0, node_bytes, stream);
    gin_scatter<<<scat_blocks, 256, 0, stream>>>(x, src, dst, agg, edge_work);
    gin_wmma_gemm<<<gemm_blocks, 256, 0, stream>>>(x, agg, W1, b1, h1, n_nodes, n_tiles, 1);

    // Layer 2
    hipMemsetAsync(agg, 0, node_bytes, stream);
    gin_scatter<<<scat_blocks, 256, 0, stream>>>(h1, src, dst, agg, edge_work);
    gin_wmma_gemm<<<gemm_blocks, 256, 0, stream>>>(h1, agg, W2, b2, h2, n_nodes, n_tiles, 1);

    // Pool + head
    hipMemsetAsync(sums, 0, pool_bytes, stream);
    gin_pool<<<pool_blocks, 256, 0, stream>>>(h2, batch, sums, cnt, node_work);
    gin_head<<<head_blocks, 256, 0, stream>>>(sums, cnt, W3, b3, out, n_grp, n_cls);
}